// SpecCnn1d_v02_20529943675596
// MI455X (gfx1250) — compile-verified
//
#include <hip/hip_runtime.h>
#include <hip/hip_bf16.h>

typedef __attribute__((ext_vector_type(2))) float v2f;
typedef __attribute__((ext_vector_type(8))) float v8f;

// Problem constants (from reference setup_inputs)
#define BSZ 128
#define ISZ 8192
#define FSZ 64
#define KSZ 3
#define OSZ 8190   // (I - K)/1 + 1
#define OCHUNK 32  // o-positions per wave iteration chunk

// out[b,o,f] = relu( sum_k phi[f,k] * (lam_in[f,o+k] - lam_out[f,o]) * x[b,o+k] )
//
// Per o: (16b x 4k) A-tile  X  (4k x 16f) B-tile  ->  (16b x 16f) D-tile
// via V_WMMA_F32_16X16X4_F32 (k=3 padded with zeros, arithmetically masked —
// no EXEC divergence anywhere in the loop).
//
// Block = 128 threads = 4 waves; wave w owns f-tile w (FSZ/16 == 4).
// grid.y = b-tile (BSZ/16 == 8), grid.x = o-chunk.
__global__ __launch_bounds__(128) void SpecCnn1d_wmma_kernel(
    const float* __restrict__ x,        // (B, I)
    const float* __restrict__ phi,      // (F, 3)
    const float* __restrict__ lam_in,   // (F, I)
    const float* __restrict__ lam_out,  // (F, O)
    float* __restrict__ out)            // (B, O, F)
{
    const int lane = threadIdx.x & 31;
    const int wid  = threadIdx.x >> 5;   // f-tile index 0..3
    const int half = lane >> 4;          // 0: lanes 0-15, 1: lanes 16-31
    const int m    = lane & 15;          // row/col-in-tile index

    const int b0 = blockIdx.y * 16;      // batch tile base
    const int f  = wid * 16 + m;         // this lane's filter column
    const int o0 = blockIdx.x * OCHUNK;

    // Per-lane invariant pointers / weights.
    // A-matrix lane roles: half0 -> k={0,1}, half1 -> k={2,pad}.
    const float* xr = x       + (size_t)(b0 + m) * ISZ;  // x row for batch b0+m
    const float* li = lam_in  + (size_t)f * ISZ;
    const float* lo = lam_out + (size_t)f * OSZ;
    const float p0    = phi[f * KSZ + (half ? 2 : 0)];
    const float p1    = half ? 0.0f : phi[f * KSZ + 1];  // zero kills the k=3 pad row of B
    const float hmask = half ? 0.0f : 1.0f;              // zero kills the k=3 pad col of A

    int o_end = o0 + OCHUNK;
    if (o_end > OSZ) o_end = OSZ;        // wave-uniform guard (EXEC stays all-1s)

    #pragma unroll 2
    for (int o = o0; o < o_end; ++o) {
        const int k0 = o + 2 * half;     // k index handled by this half (0 or 2)

        // Unconditional, always-in-range loads (o+1 <= OSZ < ISZ; k0 <= o+2 < ISZ).
        const float xa   = xr[k0];
        const float xb   = xr[o + 1];    // masked to 0 on upper half (pad k=3)
        const float lia  = li[k0];
        const float lib  = li[o + 1];
        const float lamo = lo[o];

        // A (16x4 f32): v0 = A[m, k0], v1 = A[m, k0+1] (k=3 slot arithmetically zeroed)
        v2f a;
        a.x = xa;
        a.y = hmask * xb;

        // B (4x16 f32): v0 = W[k0, n=m], v1 = W[k0+1, n=m] (p1==0 on upper half)
        v2f bm;
        bm.x = p0 * (lia - lamo);
        bm.y = p1 * (lib - lamo);

        v8f c = {};
        // 8 args: (neg_a, A, neg_b, B, c_mod, C, reuse_a, reuse_b)
        c = __builtin_amdgcn_wmma_f32_16x16x4_f32(
                /*neg_a=*/false, a, /*neg_b=*/false, bm,
                /*c_mod=*/(short)0, c, /*reuse_a=*/false, /*reuse_b=*/false);

        // D (16x16 f32): vgpr r holds batch row b0 + r + 8*half, lane -> f.
        float* op = out + ((size_t)(b0 + 8 * half) * OSZ + (size_t)o) * FSZ + f;
        #pragma unroll
        for (int r = 0; r < 8; ++r) {
            op[(size_t)r * OSZ * FSZ] = fmaxf(c[r], 0.0f);
        }
    }
}

extern "C" void kernel_launch(void* const* d_in, const int* in_sizes, int n_in,
                              void* d_out, int out_size, void* d_ws, size_t ws_size,
                              hipStream_t stream) {
    const float* x       = (const float*)d_in[0];  // (128, 8192)
    const float* phi     = (const float*)d_in[1];  // (64, 3)
    const float* lam_in  = (const float*)d_in[2];  // (64, 8192)
    const float* lam_out = (const float*)d_in[3];  // (64, 8190)
    float* out = (float*)d_out;                    // (128, 8190, 64)

    dim3 grid((OSZ + OCHUNK - 1) / OCHUNK, BSZ / 16);  // (256, 8)
    dim3 block(128);                                   // 4 waves, one per f-tile
    SpecCnn1d_wmma_kernel<<<grid, block, 0, stream>>>(x, phi, lam_in, lam_out, out);
}